// FASTopic_19121194402040
// MI455X (gfx1250) — compile-verified
//
#include <hip/hip_runtime.h>
#include <hip/hip_bf16.h>

typedef __attribute__((ext_vector_type(2))) float v2f;
typedef __attribute__((ext_vector_type(8))) float v8f;

#define EPS_OT  1e-16f
#define EPS_LOG 1e-12f

#define NDOCS_C  2048
#define VOCAB_C  30000
#define TOPICS_C 100
#define EMBED_C  200
#define KD       100          // recon inner dim (= TOPICS)
#define TWB      32           // blocks cooperating on the TW Sinkhorn
#define SINK_T   256          // threads per block in the sinkhorn kernel

// ---------------------------------------------------------------------------
// init: zero loss accumulators + inter-block barrier slots (ws not re-poisoned
// between graph replays, so this must run every call)
// ---------------------------------------------------------------------------
__global__ void k_init(float* acc, int* bar) {
    int i = blockIdx.x * blockDim.x + threadIdx.x;
    if (i < 4) acc[i] = 0.0f;
    if (i < 256) bar[i] = 0;
}

// out[r] = ||X[r,:]||^2
__global__ void k_rownorm(const float* __restrict__ X, float* __restrict__ out,
                          int rows, int d) {
    int r = blockIdx.x * blockDim.x + threadIdx.x;
    if (r >= rows) return;
    float s = 0.0f;
    const float* xr = X + (size_t)r * d;
    for (int k = 0; k < d; ++k) s += xr[k] * xr[k];
    out[r] = s;
}

// single-block softmax over n elements (blockDim = 256, power of two)
__global__ void k_softmax1b(const float* __restrict__ x, float* __restrict__ y, int n) {
    __shared__ float sd[256];
    __shared__ float smax, ssum;
    int tid = threadIdx.x, bd = blockDim.x;
    float m = -3.402823466e38f;
    for (int i = tid; i < n; i += bd) m = fmaxf(m, x[i]);
    sd[tid] = m; __syncthreads();
    for (int off = bd >> 1; off > 0; off >>= 1) {
        if (tid < off) sd[tid] = fmaxf(sd[tid], sd[tid + off]);
        __syncthreads();
    }
    if (tid == 0) smax = sd[0];
    __syncthreads();
    float s = 0.0f;
    for (int i = tid; i < n; i += bd) s += expf(x[i] - smax);
    sd[tid] = s; __syncthreads();
    for (int off = bd >> 1; off > 0; off >>= 1) {
        if (tid < off) sd[tid] += sd[tid + off];
        __syncthreads();
    }
    if (tid == 0) ssum = sd[0];
    __syncthreads();
    for (int i = tid; i < n; i += bd) y[i] = expf(x[i] - smax) / ssum;
}

// K[i] = exp(-alpha * M[i])
__global__ void k_gibbs(const float* __restrict__ M, float* __restrict__ K,
                        size_t n, float alpha) {
    size_t i = (size_t)blockIdx.x * blockDim.x + threadIdx.x;
    if (i < n) K[i] = expf(-alpha * M[i]);
}

// ---------------------------------------------------------------------------
// WMMA pairwise squared-Euclidean distance:
//   M[r,c] = xn[r] + yn[c] - 2 * dot(X[r,:], Y[c,:])
// One 16x16 tile per wave; K stepped by 4 via v_wmma_f32_16x16x4_f32.
// A frag: lanes 0-15 -> M rows, v0=K even / v1=K odd, lanes 16-31 -> K+2.
// ---------------------------------------------------------------------------
__global__ void k_pairwise_wmma(const float* __restrict__ X, const float* __restrict__ Y,
                                const float* __restrict__ xn, const float* __restrict__ yn,
                                float* __restrict__ M, int nrows, int ncols, int d) {
    const int lane = threadIdx.x & 31;
    const int wid  = threadIdx.x >> 5;
    const int tiles_c = (ncols + 15) >> 4;
    const int tiles_r = (nrows + 15) >> 4;
    int tile = blockIdx.x * (blockDim.x >> 5) + wid;   // wave-uniform
    if (tile >= tiles_c * tiles_r) return;
    const int tr = tile / tiles_c, tc = tile % tiles_c;
    const int row0 = tr << 4, col0 = tc << 4;
    const int half = lane >> 4;
    const int l16  = lane & 15;
    int ra = row0 + l16; if (ra >= nrows) ra = nrows - 1;   // clamp, mask on store
    int cb = col0 + l16; if (cb >= ncols) cb = ncols - 1;
    const float* xr = X + (size_t)ra * d;
    const float* yr = Y + (size_t)cb * d;
    v8f acc = {};
    for (int kk = 0; kk < d; kk += 4) {
        const int ka = kk + 2 * half;
        v2f a, b;
        a.x = xr[ka]; a.y = xr[ka + 1];
        b.x = yr[ka]; b.y = yr[ka + 1];
        acc = __builtin_amdgcn_wmma_f32_16x16x4_f32(false, a, false, b,
                                                    (short)0, acc, false, false);
    }
    const int c = col0 + l16;
    if (c < ncols) {
        const float ync = yn[c];
        for (int i = 0; i < 8; ++i) {
            const int r = row0 + i + 8 * half;
            if (r < nrows)
                M[(size_t)r * ncols + c] = xn[r] + ync - 2.0f * acc[i];
        }
    }
}

// ---------------------------------------------------------------------------
// Persistent Sinkhorn: ONE launch runs all 100 iterations of BOTH OT problems.
//   blocks 0..TWB-1 : TW problem (100 x 30000), inter-block slot barrier
//   block  TWB      : DT problem (2048 x 100), LDS-resident u/v, __syncthreads
// Reference order per iteration: v = b/(K^T u + eps) ; u = a/(K v + eps).
// ---------------------------------------------------------------------------
__device__ inline void grid_barrier(int* bar, int slot, int nblk) {
    __syncthreads();
    if (threadIdx.x == 0) {
        __threadfence();
        atomicAdd(&bar[slot], 1);
        volatile int* p = bar + slot;
        while (*p < nblk) { __builtin_amdgcn_s_sleep(1); }
        __threadfence();
    }
    __syncthreads();
}

__global__ void k_sinkhorn(const float* __restrict__ K_DT, const float* __restrict__ bDT,
                           float* __restrict__ uDT, float* __restrict__ vDT,
                           const float* __restrict__ K_TW, const float* __restrict__ bTW,
                           float* __restrict__ uTW, float* __restrict__ vTW,
                           float* __restrict__ sTW, int* __restrict__ bar, int iters) {
    const float aDT = 1.0f / (float)NDOCS_C;
    const float aTW = 1.0f / (float)TOPICS_C;
    const int tid = threadIdx.x, wid = tid >> 5, lane = tid & 31;

    __shared__ float uL[NDOCS_C];     // DT block: u resident (8 KB of 320 KB LDS)
    __shared__ float vL[TOPICS_C + 4];
    __shared__ float uTl[TOPICS_C + 4];

    if (blockIdx.x < TWB) {
        // ---------------- TW problem: n=100 rows, m=30000 cols ----------------
        const int V = VOCAB_C, T = TOPICS_C;
        const int chunk = (V + TWB - 1) / TWB;                 // 938
        const int j0 = blockIdx.x * chunk;
        const int j1 = (j0 + chunk < V) ? j0 + chunk : V;
        for (int it = 0; it < iters; ++it) {
            // u_{it-1} recomputed from previous row-sums (u0 = 1/T)
            if (tid < T) uTl[tid] = (it == 0) ? (1.0f / T) : (aTW / (sTW[tid] + EPS_OT));
            __syncthreads();
            // v-step over this block's columns
            for (int j = j0 + tid; j < j1; j += blockDim.x) {
                float s = 0.0f;
                for (int i = 0; i < T; ++i) s += K_TW[(size_t)i * V + j] * uTl[i];
                vTW[j] = bTW[j] / (s + EPS_OT);
            }
            grid_barrier(bar, 2 * it, TWB);
            // row-sums s[i] = sum_j K[i,j] v[j] : one wave per row, shuffle-reduced
            const int gw = blockIdx.x * (SINK_T >> 5) + wid;   // 0..255 global wave
            if (gw < T) {
                const float* kr = K_TW + (size_t)gw * V;
                float s = 0.0f;
                for (int j = lane; j < V; j += 32) s += kr[j] * vTW[j];
                for (int off = 16; off > 0; off >>= 1) s += __shfl_xor(s, off, 32);
                if (lane == 0) sTW[gw] = s;
            }
            grid_barrier(bar, 2 * it + 1, TWB);
        }
        if (blockIdx.x == 0 && tid < TOPICS_C)                 // final u
            uTW[tid] = aTW / (sTW[tid] + EPS_OT);
    } else {
        // ---------------- DT problem: n=2048 rows, m=100 cols -----------------
        const int n = NDOCS_C, m = TOPICS_C;
        for (int i = tid; i < n; i += blockDim.x) uL[i] = 1.0f / n;
        __syncthreads();
        for (int it = 0; it < iters; ++it) {
            // v-step: wave w owns columns w, w+8, ...; lanes stride rows
            for (int j = wid; j < m; j += (SINK_T >> 5)) {
                float s = 0.0f;
                for (int i = lane; i < n; i += 32) s += K_DT[(size_t)i * m + j] * uL[i];
                for (int off = 16; off > 0; off >>= 1) s += __shfl_xor(s, off, 32);
                if (lane == 0) vL[j] = bDT[j] / (s + EPS_OT);
            }
            __syncthreads();
            // u-step
            for (int i = tid; i < n; i += blockDim.x) {
                const float* kr = K_DT + (size_t)i * m;
                float s = 0.0f;
                for (int j = 0; j < m; ++j) s += kr[j] * vL[j];
                uL[i] = aDT / (s + EPS_OT);
            }
            __syncthreads();
        }
        for (int i = tid; i < n; i += blockDim.x) uDT[i] = uL[i];
        if (tid < m) vDT[tid] = vL[tid];
    }
}

// loss += sum_{i,j} u[i]*K[i,j]*v[j]*M[i,j]
__global__ void k_transp_loss(const float* __restrict__ K, const float* __restrict__ M,
                              const float* __restrict__ u, const float* __restrict__ v,
                              float* __restrict__ acc, int n, int m) {
    __shared__ float sd[256];
    const size_t total = (size_t)n * m;
    float s = 0.0f;
    for (size_t idx = (size_t)blockIdx.x * blockDim.x + threadIdx.x;
         idx < total; idx += (size_t)gridDim.x * blockDim.x) {
        const int i = (int)(idx / m), j = (int)(idx % m);
        s += u[i] * K[idx] * v[j] * M[idx];
    }
    sd[threadIdx.x] = s; __syncthreads();
    for (int off = blockDim.x >> 1; off > 0; off >>= 1) {
        if (threadIdx.x < off) sd[threadIdx.x] += sd[threadIdx.x + off];
        __syncthreads();
    }
    if (threadIdx.x == 0) atomicAdd(acc, sd[0]);
}

// theta[i,j] = u[i]*K[i,j]*v[j]*scale    (row-major, [n,m])
__global__ void k_transp_scale(const float* __restrict__ K, const float* __restrict__ u,
                               const float* __restrict__ v, float scale,
                               float* __restrict__ out, int n, int m) {
    const size_t total = (size_t)n * m;
    const size_t idx = (size_t)blockIdx.x * blockDim.x + threadIdx.x;
    if (idx >= total) return;
    const int i = (int)(idx / m), j = (int)(idx % m);
    out[idx] = u[i] * K[idx] * v[j] * scale;
}

// betaT[j,i] = u[i]*K[i,j]*v[j]*scale    (TRANSPOSED, [m,n] so recon B-frags
// become contiguous 8B-aligned pairs -> global_load_b64)
__global__ void k_transp_scale_T(const float* __restrict__ K, const float* __restrict__ u,
                                 const float* __restrict__ v, float scale,
                                 float* __restrict__ outT, int n, int m) {
    const size_t total = (size_t)n * m;
    const size_t idx = (size_t)blockIdx.x * blockDim.x + threadIdx.x;
    if (idx >= total) return;
    const int i = (int)(idx / m), j = (int)(idx % m);
    outT[(size_t)j * n + i] = u[i] * K[idx] * v[j] * scale;
}

// ---------------------------------------------------------------------------
// Fused recon + DSR: acc += sum bow * log(theta@beta + eps), recon never
// materialized. Each wave caches all 25 B-fragments (one 16-col tile of
// betaT, 50 VGPRs) and reuses them across 4 row tiles; bow streams once from
// HBM (the only mandatory 245 MB of traffic), with prefetch of the next tile.
// ---------------------------------------------------------------------------
__global__ void k_recon_dsr_wmma(const float* __restrict__ theta,   // [N, KD]
                                 const float* __restrict__ betaT,   // [V, KD]
                                 const float* __restrict__ bow,     // [N, V]
                                 float* __restrict__ acc, int N, int V) {
    const int lane = threadIdx.x & 31;
    const int wid  = threadIdx.x >> 5;
    const int tiles_v = V >> 4;                    // 1875
    const int rgroups = N >> 6;                    // 4 row-tiles per wave: 32
    long wtile = (long)blockIdx.x * (blockDim.x >> 5) + wid;   // wave-uniform
    const long nwt = (long)rgroups * tiles_v;      // 60000
    if (wtile >= nwt) return;
    const int rg = (int)(wtile / tiles_v), tv = (int)(wtile % tiles_v);
    const int col0 = tv << 4;
    const int half = lane >> 4, l16 = lane & 15;
    const int cidx = col0 + l16;

    v2f bf[KD / 4];
    const float* br = betaT + (size_t)cidx * KD;
#pragma unroll
    for (int s = 0; s < KD / 4; ++s) {
        bf[s].x = br[4 * s + 2 * half];
        bf[s].y = br[4 * s + 2 * half + 1];
    }

    float ssum = 0.0f;
    for (int rt = 0; rt < 4; ++rt) {
        const int row0 = (rg * 4 + rt) << 4;
        if (rt < 3)  // prefetch next tile's bow lines (global_prefetch_b8)
            __builtin_prefetch(&bow[(size_t)(row0 + 16 + l16) * V + cidx], 0, 1);
        const float* ar = theta + (size_t)(row0 + l16) * KD;
        v8f c = {};
#pragma unroll
        for (int s = 0; s < KD / 4; ++s) {
            v2f a;
            a.x = ar[4 * s + 2 * half];
            a.y = ar[4 * s + 2 * half + 1];
            c = __builtin_amdgcn_wmma_f32_16x16x4_f32(false, a, false, bf[s],
                                                      (short)0, c, false, false);
        }
#pragma unroll
        for (int i = 0; i < 8; ++i) {
            const int r = row0 + i + 8 * half;
            ssum += bow[(size_t)r * V + cidx] * logf(c[i] + EPS_LOG);
        }
    }
    for (int off = 16; off > 0; off >>= 1) ssum += __shfl_xor(ssum, off, 32);
    if (lane == 0) atomicAdd(acc, ssum);
}

// out[0]=loss, out[1]=loss_DSR, out[2]=loss_ETP
__global__ void k_finalize(const float* __restrict__ acc, float* __restrict__ out,
                           float invN) {
    const float etp = acc[0] + acc[1];
    const float dsr = -acc[2] * invN;
    out[0] = dsr + etp;
    out[1] = dsr;
    out[2] = etp;
}

// ---------------------------------------------------------------------------
// Launch
// ---------------------------------------------------------------------------
extern "C" void kernel_launch(void* const* d_in, const int* in_sizes, int n_in,
                              void* d_out, int out_size, void* d_ws, size_t ws_size,
                              hipStream_t stream) {
    (void)in_sizes; (void)n_in; (void)out_size; (void)ws_size;
    const int NDOCS = NDOCS_C, VOCAB = VOCAB_C, TOPICS = TOPICS_C, EMBED = EMBED_C;
    const float DT_ALPHA = 3.0f, TW_ALPHA = 2.0f;
    const int ITERS = 100;

    const float* bow  = (const float*)d_in[0];   // [2048,30000]
    const float* docE = (const float*)d_in[1];   // [2048,200]
    const float* wrdE = (const float*)d_in[2];   // [30000,200]
    const float* topE = (const float*)d_in[3];   // [100,200]
    const float* wrdW = (const float*)d_in[4];   // [30000]
    const float* topW = (const float*)d_in[5];   // [100]
    float* out = (float*)d_out;

    // workspace carve-up (floats)
    float* ws = (float*)d_ws;
    size_t cur = 0;
    auto take = [&](size_t n) { float* p = ws + cur; cur += (n + 63) & ~(size_t)63; return p; };
    float* acc   = take(4);                      // [0]=lossDT [1]=lossTW [2]=dsr_sum
    int*   bar   = (int*)take(256);              // inter-block barrier slots
    float* uDT   = take(NDOCS);
    float* vDT   = take(TOPICS);
    float* bDT   = take(TOPICS);
    float* uTW   = take(TOPICS);
    float* vTW   = take(VOCAB);
    float* bTW   = take(VOCAB);
    float* sTW   = take(TOPICS);
    float* docN  = take(NDOCS);
    float* topN  = take(TOPICS);
    float* wrdN  = take(VOCAB);
    float* M_DT  = take((size_t)NDOCS * TOPICS);
    float* K_DT  = take((size_t)NDOCS * TOPICS);
    float* theta = take((size_t)NDOCS * TOPICS);
    float* M_TW  = take((size_t)TOPICS * VOCAB);
    float* K_TW  = take((size_t)TOPICS * VOCAB);
    float* betaT = take((size_t)TOPICS * VOCAB); // transposed [V, TOPICS]

    // 1) zero accumulators + barrier slots (must happen every call)
    k_init<<<1, 256, 0, stream>>>(acc, bar);

    // 2) row squared norms
    k_rownorm<<<(NDOCS + 255) / 256, 256, 0, stream>>>(docE, docN, NDOCS, EMBED);
    k_rownorm<<<(TOPICS + 255) / 256, 256, 0, stream>>>(topE, topN, TOPICS, EMBED);
    k_rownorm<<<(VOCAB + 255) / 256, 256, 0, stream>>>(wrdE, wrdN, VOCAB, EMBED);

    // 3) marginals b = softmax(weights)
    k_softmax1b<<<1, 256, 0, stream>>>(topW, bDT, TOPICS);
    k_softmax1b<<<1, 256, 0, stream>>>(wrdW, bTW, VOCAB);

    // 4) cost matrices via WMMA f32 16x16x4, then Gibbs kernels
    {
        int tiles = ((NDOCS + 15) / 16) * ((TOPICS + 15) / 16);      // 128*7
        k_pairwise_wmma<<<(tiles + 3) / 4, 128, 0, stream>>>(docE, topE, docN, topN,
                                                             M_DT, NDOCS, TOPICS, EMBED);
        size_t n = (size_t)NDOCS * TOPICS;
        k_gibbs<<<(int)((n + 255) / 256), 256, 0, stream>>>(M_DT, K_DT, n, DT_ALPHA);
    }
    {
        int tiles = ((TOPICS + 15) / 16) * ((VOCAB + 15) / 16);      // 7*1875
        k_pairwise_wmma<<<(tiles + 3) / 4, 128, 0, stream>>>(topE, wrdE, topN, wrdN,
                                                             M_TW, TOPICS, VOCAB, EMBED);
        size_t n = (size_t)TOPICS * VOCAB;
        k_gibbs<<<(int)((n + 255) / 256), 256, 0, stream>>>(M_TW, K_TW, n, TW_ALPHA);
    }

    // 5) persistent Sinkhorn: ONE launch, both problems, all 100 iterations
    k_sinkhorn<<<TWB + 1, SINK_T, 0, stream>>>(K_DT, bDT, uDT, vDT,
                                               K_TW, bTW, uTW, vTW,
                                               sTW, bar, ITERS);

    // 6) OT losses: sum(transp * M)
    k_transp_loss<<<512, 256, 0, stream>>>(K_DT, M_DT, uDT, vDT, acc + 0, NDOCS, TOPICS);
    k_transp_loss<<<2048, 256, 0, stream>>>(K_TW, M_TW, uTW, vTW, acc + 1, TOPICS, VOCAB);

    // 7) theta = transp_DT * N (row-major) ; betaT = (transp_TW * K)^T
    {
        size_t n = (size_t)NDOCS * TOPICS;
        k_transp_scale<<<(int)((n + 255) / 256), 256, 0, stream>>>(K_DT, uDT, vDT,
                                                                   (float)NDOCS, theta,
                                                                   NDOCS, TOPICS);
        size_t m = (size_t)TOPICS * VOCAB;
        k_transp_scale_T<<<(int)((m + 255) / 256), 256, 0, stream>>>(K_TW, uTW, vTW,
                                                                     (float)TOPICS, betaT,
                                                                     TOPICS, VOCAB);
    }

    // 8) fused recon + DSR (streams bow once; 100 WMMAs per wave, B-frags cached)
    {
        long nwt = (long)(NDOCS / 64) * (VOCAB / 16);                // 60000 waves
        int blocks = (int)((nwt + 3) / 4);
        k_recon_dsr_wmma<<<blocks, 128, 0, stream>>>(theta, betaT, bow, acc + 2,
                                                     NDOCS, VOCAB);
    }

    // 9) finalize three scalar outputs
    k_finalize<<<1, 1, 0, stream>>>(acc, out, 1.0f / (float)NDOCS);
}